// CMTLiDAR_74672301408423
// MI455X (gfx1250) — compile-verified
//
#include <hip/hip_runtime.h>

typedef float v2f __attribute__((ext_vector_type(2)));
typedef float v8f __attribute__((ext_vector_type(8)));

#define BATCH   2
#define NQ      4096
#define CDIM    512
#define TOPK_N  120
#define MROWS   (BATCH * NQ)

#define WAVES        4
#define ROWS_PER_WG  (WAVES * 16)
#define ASTRIDE      (CDIM + 4)   // +4 floats pad: rows land on distinct LDS bank groups

// workspace layout (float offsets)
#define WS_CENTERS  0            // MROWS*3 = 24576
#define WS_SCORE    24576        // MROWS   =  8192
#define WS_TIDX     32768        // 240 ints
#define WS_TSCORE   33008        // 240 floats
#define WS_BT       33280        // 3*256*512 float2 = 786432 floats (8B aligned)
#define WS_END      (WS_BT + 3 * 256 * CDIM * 2)

__device__ __forceinline__ float redux16(float x) {
  // sum across the 16 lanes of each half-wave (xor masks 1,2,4,8 stay inside halves)
  x += __shfl_xor(x, 1, 32);
  x += __shfl_xor(x, 2, 32);
  x += __shfl_xor(x, 4, 32);
  x += __shfl_xor(x, 8, 32);
  return x;
}

__device__ __forceinline__ unsigned long long umax64(unsigned long long a, unsigned long long b) {
  return a > b ? a : b;
}

__device__ __forceinline__ float sigmoidf(float x) { return 1.0f / (1.0f + expf(-x)); }

__device__ __forceinline__ float inv_sigmoidf(float p) {
  p = fminf(fmaxf(p, 1e-5f), 1.0f - 1e-5f);
  return logf(p / (1.0f - p));
}

// ---------------------------------------------------------------------------
// Kernel 0 (packed path): repack Wh[h][k][col] into Bt[h][t][col] float2 pairs
// (Wh[h][2t][col], Wh[h][2t+1][col]) so a B fragment half is one b64 load.
// ---------------------------------------------------------------------------
__global__ __launch_bounds__(256) void k_pack(const float* __restrict__ Wh,
                                              float2* __restrict__ Bt) {
  const int idx = blockIdx.x * 256 + threadIdx.x;   // ((h*256 + t)*512 + col)
  if (idx >= 3 * 256 * CDIM) return;
  const int col  = idx & (CDIM - 1);
  const int rest = idx >> 9;
  const int t    = rest & 255;
  const int h    = rest >> 8;
  const float* base = Wh + ((size_t)h * CDIM + 2 * t) * CDIM + col;
  float2 v; v.x = base[0]; v.y = base[CDIM];
  Bt[idx] = v;
}

// ---------------------------------------------------------------------------
// Kernel 1: last-layer 3-head MLP via fp32 WMMA, fused small heads + epilogue.
// Produces centers[MROWS,3] and score[MROWS] (class-0 softmax prob).
// PACKED=true  -> B fragments from repacked Bt (1 b64 per wmma)
// PACKED=false -> B fragments straight from Wh (2 b32 per wmma)
// ---------------------------------------------------------------------------
template <bool PACKED>
__global__ __launch_bounds__(WAVES * 32) void k_heads(
    const float* __restrict__ Alast,   // outs_dec[-1] : [MROWS, CDIM]
    const float* __restrict__ Wh,      // [3, CDIM, CDIM]
    const float* __restrict__ Bt,      // packed pairs (or nullptr)
    const float* __restrict__ bh,      // [3, CDIM]
    const float* __restrict__ Wc,      // [CDIM, 2]
    const float* __restrict__ bc,      // [2]
    const float* __restrict__ Whh,     // [CDIM, 1]
    const float* __restrict__ bhh,     // [1]
    const float* __restrict__ Wcls,    // [CDIM, 3]
    const float* __restrict__ bcls,    // [3]
    const float* __restrict__ refpts,  // [MROWS, 3]
    float* __restrict__ centers,       // [MROWS, 3]
    float* __restrict__ score)         // [MROWS]
{
  extern __shared__ float smem[];
  float* Atile  = smem;                              // WAVES*16*ASTRIDE
  float* rowout = smem + WAVES * 16 * ASTRIDE;       // ROWS_PER_WG * 8

  const int tid  = threadIdx.x;
  const int wave = tid >> 5;
  const int lane = tid & 31;
  const int l16  = lane & 15;
  const int half = lane >> 4;                        // 0: lanes 0-15, 1: lanes 16-31
  const int wg_row0   = blockIdx.x * ROWS_PER_WG;
  const int wave_row0 = wg_row0 + wave * 16;

  // ---- stage this wave's 16x512 fp32 A tile into LDS (float4 coalesced) ----
  for (int t = lane; t < 16 * 128; t += 32) {
    const int row = t >> 7;        // 0..15
    const int kq  = t & 127;       // float4 index 0..127
    const float4 v = ((const float4*)(Alast + (size_t)(wave_row0 + row) * CDIM))[kq];
    *(float4*)&Atile[(wave * 16 + row) * ASTRIDE + (kq << 2)] = v;
  }
  __syncthreads();

  // A fragment (16x4 f32): VGPR0 = {lanes0-15: K=k0, lanes16-31: K=k0+2}, VGPR1 = +1
  const int   krow = half * 2;
  const float* aptr = &Atile[(wave * 16 + l16) * ASTRIDE + krow];

  // per-row small-head accumulators: rows m = v + 8*half of this wave's tile
  float cen0[8], cen1[8], hei[8], cl0[8], cl1[8], cl2[8];
  #pragma unroll
  for (int v = 0; v < 8; ++v) { cen0[v] = cen1[v] = hei[v] = cl0[v] = cl1[v] = cl2[v] = 0.0f; }

  #pragma unroll 1
  for (int h = 0; h < 3; ++h) {
    #pragma unroll 1
    for (int jg = 0; jg < 32; jg += 4) {               // 4 hidden-col chunks per K-sweep
      v8f acc[4] = {};

      if (PACKED) {
        // Bt[h][t][col], t = k0/2 + half; one b64 per wmma operand
        const v2f* bp0 = (const v2f*)Bt + ((size_t)h * 256 + half) * CDIM + ((jg + 0) * 16 + l16);
        const v2f* bp1 = bp0 + 16;
        const v2f* bp2 = bp0 + 32;
        const v2f* bp3 = bp0 + 48;
        #pragma unroll 4
        for (int k0 = 0; k0 < CDIM; k0 += 4) {
          v2f a; a.x = aptr[k0]; a.y = aptr[k0 + 1];
          const v2f b0 = bp0[0], b1 = bp1[0], b2 = bp2[0], b3 = bp3[0];
          acc[0] = __builtin_amdgcn_wmma_f32_16x16x4_f32(false, a, false, b0, (short)0, acc[0], false, false);
          acc[1] = __builtin_amdgcn_wmma_f32_16x16x4_f32(false, a, false, b1, (short)0, acc[1], false, false);
          acc[2] = __builtin_amdgcn_wmma_f32_16x16x4_f32(false, a, false, b2, (short)0, acc[2], false, false);
          acc[3] = __builtin_amdgcn_wmma_f32_16x16x4_f32(false, a, false, b3, (short)0, acc[3], false, false);
          bp0 += 2 * CDIM; bp1 += 2 * CDIM; bp2 += 2 * CDIM; bp3 += 2 * CDIM;
        }
      } else {
        const float* Bw = Wh + (size_t)h * CDIM * CDIM;
        const float* bp0 = Bw + (size_t)krow * CDIM + ((jg + 0) * 16 + l16);
        const float* bp1 = bp0 + 16;
        const float* bp2 = bp0 + 32;
        const float* bp3 = bp0 + 48;
        #pragma unroll 4
        for (int k0 = 0; k0 < CDIM; k0 += 4) {
          v2f a; a.x = aptr[k0]; a.y = aptr[k0 + 1];
          v2f b0; b0.x = bp0[0]; b0.y = bp0[CDIM];
          v2f b1; b1.x = bp1[0]; b1.y = bp1[CDIM];
          v2f b2; b2.x = bp2[0]; b2.y = bp2[CDIM];
          v2f b3; b3.x = bp3[0]; b3.y = bp3[CDIM];
          acc[0] = __builtin_amdgcn_wmma_f32_16x16x4_f32(false, a, false, b0, (short)0, acc[0], false, false);
          acc[1] = __builtin_amdgcn_wmma_f32_16x16x4_f32(false, a, false, b1, (short)0, acc[1], false, false);
          acc[2] = __builtin_amdgcn_wmma_f32_16x16x4_f32(false, a, false, b2, (short)0, acc[2], false, false);
          acc[3] = __builtin_amdgcn_wmma_f32_16x16x4_f32(false, a, false, b3, (short)0, acc[3], false, false);
          bp0 += 4 * CDIM; bp1 += 4 * CDIM; bp2 += 4 * CDIM; bp3 += 4 * CDIM;
        }
      }

      // bias + ReLU, fold hidden chunk straight into tiny head outputs.
      // C layout: VGPR v holds (M=v, N=l16) for lanes0-15, (M=v+8, N=l16) for lanes16-31.
      #pragma unroll
      for (int q = 0; q < 4; ++q) {
        const int col  = (jg + q) * 16 + l16;
        const float bias = bh[h * CDIM + col];
        float w0 = 0.f, w1 = 0.f, w2 = 0.f;
        if (h == 0)      { w0 = Wc[col * 2 + 0]; w1 = Wc[col * 2 + 1]; }
        else if (h == 1) { w0 = Whh[col]; }
        else             { w0 = Wcls[col * 3 + 0]; w1 = Wcls[col * 3 + 1]; w2 = Wcls[col * 3 + 2]; }
        #pragma unroll
        for (int v = 0; v < 8; ++v) {
          float hv = fmaxf(acc[q][v] + bias, 0.0f);
          if (h == 0)      { cen0[v] += hv * w0; cen1[v] += hv * w1; }
          else if (h == 1) { hei[v]  += hv * w0; }
          else             { cl0[v] += hv * w0; cl1[v] += hv * w1; cl2[v] += hv * w2; }
        }
      }
    }
  }

  // reduce the N (column) dimension across each 16-lane group
  #pragma unroll
  for (int v = 0; v < 8; ++v) {
    cen0[v] = redux16(cen0[v]); cen1[v] = redux16(cen1[v]); hei[v] = redux16(hei[v]);
    cl0[v]  = redux16(cl0[v]);  cl1[v]  = redux16(cl1[v]);  cl2[v] = redux16(cl2[v]);
  }
  if (l16 == 0) {
    #pragma unroll
    for (int v = 0; v < 8; ++v) {
      const int lr = wave * 16 + half * 8 + v;
      rowout[lr * 8 + 0] = cen0[v]; rowout[lr * 8 + 1] = cen1[v]; rowout[lr * 8 + 2] = hei[v];
      rowout[lr * 8 + 3] = cl0[v];  rowout[lr * 8 + 4] = cl1[v];  rowout[lr * 8 + 5] = cl2[v];
    }
  }
  __syncthreads();

  // per-row epilogue: inverse-sigmoid ref, sigmoid + pc_range scale, softmax class-0
  if (tid < ROWS_PER_WG) {
    const int gr = wg_row0 + tid;
    const float c0 = rowout[tid * 8 + 0] + bc[0];
    const float c1 = rowout[tid * 8 + 1] + bc[1];
    const float hh = rowout[tid * 8 + 2] + bhh[0];
    const float l0 = rowout[tid * 8 + 3] + bcls[0];
    const float l1 = rowout[tid * 8 + 4] + bcls[1];
    const float l2 = rowout[tid * 8 + 5] + bcls[2];
    const float r0 = inv_sigmoidf(refpts[gr * 3 + 0]);
    const float r1 = inv_sigmoidf(refpts[gr * 3 + 1]);
    const float r2 = inv_sigmoidf(refpts[gr * 3 + 2]);
    const float cx = sigmoidf(c0 + r0) * 108.0f - 54.0f;   // pc_range x: [-54, 54]
    const float cy = sigmoidf(c1 + r1) * 108.0f - 54.0f;   // pc_range y: [-54, 54]
    const float hz = sigmoidf(hh + r2) * 8.0f - 5.0f;      // pc_range z: [-5, 3]
    centers[gr * 3 + 0] = cx; centers[gr * 3 + 1] = cy; centers[gr * 3 + 2] = hz;
    const float mx = fmaxf(l0, fmaxf(l1, l2));
    const float e0 = expf(l0 - mx), e1 = expf(l1 - mx), e2 = expf(l2 - mx);
    score[gr] = e0 / (e0 + e1 + e2);
  }
}

// ---------------------------------------------------------------------------
// Kernel 2: deterministic top-120 per batch via a 32-segment tournament.
// Keys = (score_bits<<32)|(4095-n): ties resolve to lowest index (jax top_k).
// After init, a single wave runs all 120 rounds with no block barriers:
// each round = 32-lane max over segment maxima + rescan of one 128-segment.
// ---------------------------------------------------------------------------
__global__ __launch_bounds__(256) void k_topk(
    const float* __restrict__ score,   // [BATCH, NQ]
    int*   __restrict__ topk_idx,      // [BATCH, TOPK_N]
    float* __restrict__ topk_score)    // [BATCH, TOPK_N]
{
  __shared__ unsigned long long keys[NQ];      // 32 KB
  __shared__ unsigned long long segmax[32];

  const int b    = blockIdx.x;
  const int tid  = threadIdx.x;
  const int lane = tid & 31;
  const int wave = tid >> 5;
  const float* s = score + (size_t)b * NQ;

  for (int i = tid; i < NQ; i += 256) {
    const unsigned int fb = __float_as_uint(s[i]);   // softmax output > 0: bits are monotone
    keys[i] = ((unsigned long long)fb << 32) | (unsigned int)(NQ - 1 - i);
  }
  __syncthreads();

  // init segment maxima: 32 segments x 128 elements, 8 threads per segment
  {
    const int seg = tid >> 3, sub = tid & 7;
    unsigned long long m = 0ull;
    #pragma unroll 4
    for (int j = 0; j < 16; ++j) m = umax64(m, keys[seg * 128 + sub * 16 + j]);
    m = umax64(m, __shfl_xor(m, 1, 32));
    m = umax64(m, __shfl_xor(m, 2, 32));
    m = umax64(m, __shfl_xor(m, 4, 32));
    if (sub == 0) segmax[seg] = m;
  }
  __syncthreads();

  if (wave == 0) {
    for (int it = 0; it < TOPK_N; ++it) {
      unsigned long long v = segmax[lane];
      #pragma unroll
      for (int off = 16; off >= 1; off >>= 1) v = umax64(v, __shfl_xor(v, off, 32));
      const int idx = (NQ - 1) - (int)(v & 0xFFFFFFFFull);
      if (lane == 0) {
        topk_idx[b * TOPK_N + it]   = idx;
        topk_score[b * TOPK_N + it] = __uint_as_float((unsigned int)(v >> 32));
        keys[idx] = 0ull;                       // remove winner (same-wave LDS order)
      }
      // recompute the winner segment's max (128 elements, 4 per lane)
      const int seg = idx >> 7;
      unsigned long long m = 0ull;
      #pragma unroll
      for (int j = 0; j < 4; ++j) m = umax64(m, keys[seg * 128 + j * 32 + lane]);
      #pragma unroll
      for (int off = 16; off >= 1; off >>= 1) m = umax64(m, __shfl_xor(m, off, 32));
      if (lane == 0) segmax[seg] = m;
    }
  }
}

// ---------------------------------------------------------------------------
// Kernel 3: gather + pack [query(512) | ref(3) | center(3) | score] per row.
// ---------------------------------------------------------------------------
__global__ __launch_bounds__(128) void k_gather(
    const float* __restrict__ Alast,     // [MROWS, CDIM]
    const float* __restrict__ refpts,    // [MROWS, 3]
    const float* __restrict__ centers,   // [MROWS, 3]
    const int*   __restrict__ topk_idx,  // [BATCH*TOPK_N]
    const float* __restrict__ topk_score,// [BATCH*TOPK_N]
    float* __restrict__ out)             // [BATCH, TOPK_N, CDIM+7]
{
  const int row = blockIdx.x;            // 0 .. BATCH*TOPK_N-1
  const int b   = row / TOPK_N;
  const int idx = topk_idx[row];
  const int gr  = b * NQ + idx;
  const float* src = Alast + (size_t)gr * CDIM;
  float* dst = out + (size_t)row * (CDIM + 7);

  for (int i = threadIdx.x; i < CDIM; i += 128) dst[i] = src[i];
  if (threadIdx.x < 3)       dst[CDIM + threadIdx.x]     = refpts[gr * 3 + threadIdx.x];
  else if (threadIdx.x < 6)  dst[CDIM + threadIdx.x]     = centers[gr * 3 + (threadIdx.x - 3)];
  else if (threadIdx.x == 6) dst[CDIM + 6]               = topk_score[row];
}

// ---------------------------------------------------------------------------
extern "C" void kernel_launch(void* const* d_in, const int* in_sizes, int n_in,
                              void* d_out, int out_size, void* d_ws, size_t ws_size,
                              hipStream_t stream) {
  const float* outs_dec = (const float*)d_in[0];   // [6, 2, 4096, 512]
  const float* refpts   = (const float*)d_in[1];   // [2, 4096, 3]
  const float* Wh       = (const float*)d_in[2];   // [3, 512, 512]
  const float* bh       = (const float*)d_in[3];   // [3, 512]
  const float* Wc       = (const float*)d_in[4];   // [512, 2]
  const float* bc       = (const float*)d_in[5];   // [2]
  const float* Whh      = (const float*)d_in[6];   // [512, 1]
  const float* bhh      = (const float*)d_in[7];   // [1]
  const float* Wcls     = (const float*)d_in[8];   // [512, 3]
  const float* bcls     = (const float*)d_in[9];   // [3]
  float* out = (float*)d_out;

  // only the last decoder layer feeds the output
  const float* Alast = outs_dec + (size_t)(6 - 1) * MROWS * CDIM;

  float* ws      = (float*)d_ws;
  float* centers = ws + WS_CENTERS;
  float* score   = ws + WS_SCORE;
  int*   tidx    = (int*)(ws + WS_TIDX);
  float* tscore  = ws + WS_TSCORE;
  float* Bt      = ws + WS_BT;

  const bool packed = ws_size >= (size_t)WS_END * sizeof(float);

  const size_t smem1 = (size_t)(WAVES * 16 * ASTRIDE + ROWS_PER_WG * 8) * sizeof(float);
  const dim3 grid1(MROWS / ROWS_PER_WG), blk1(WAVES * 32);

  if (packed) {
    k_pack<<<dim3((3 * 256 * CDIM) / 256), dim3(256), 0, stream>>>(Wh, (float2*)Bt);
    k_heads<true><<<grid1, blk1, smem1, stream>>>(
        Alast, Wh, Bt, bh, Wc, bc, Whh, bhh, Wcls, bcls, refpts, centers, score);
  } else {
    k_heads<false><<<grid1, blk1, smem1, stream>>>(
        Alast, Wh, nullptr, bh, Wc, bc, Whh, bhh, Wcls, bcls, refpts, centers, score);
  }

  k_topk<<<dim3(BATCH), dim3(256), 0, stream>>>(score, tidx, tscore);

  k_gather<<<dim3(BATCH * TOPK_N), dim3(128), 0, stream>>>(
      Alast, refpts, centers, tidx, tscore, out);
}